// FPSDownsample_26242250178592
// MI455X (gfx1250) — compile-verified
//
#include <hip/hip_runtime.h>

// MI455X / gfx1250, wave32.
// FPS: 1 persistent WG (32 waves) per batch; x (96MB) + dist (32MB) stay L2-resident.
// MLP layers 2/3: V_WMMA_F32_16X16X4_F32, weights staged into LDS via the
// Tensor Data Mover (tensor_load_to_lds, TENSORcnt) when the builtin exists.

typedef __attribute__((ext_vector_type(2))) float v2f;
typedef __attribute__((ext_vector_type(8))) float v8f;
typedef unsigned int v4u __attribute__((ext_vector_type(4)));
typedef int v4i __attribute__((ext_vector_type(4)));
typedef int v8i __attribute__((ext_vector_type(8)));

#if defined(__has_builtin)
#if __has_builtin(__builtin_amdgcn_tensor_load_to_lds)
#define HAVE_TDM 1
#endif
#endif

#define B_ 64
#define N_ 131072
#define S_ 1024
#define FPS_THREADS 1024

__device__ __forceinline__ void argmax_upd(float& bd, int& bi, float od, int oi) {
  // match jnp.argmax tie-break: first (lowest) index wins on equality
  if (od > bd || (od == bd && oi < bi)) { bd = od; bi = oi; }
}

// ---------------- FPS: one workgroup (32 waves) per batch ----------------
__global__ __launch_bounds__(FPS_THREADS) void fps_kernel(
    const float* __restrict__ x, float* __restrict__ dist,
    float* __restrict__ sampled) {
  __shared__ float sD[32];
  __shared__ int   sI[32];
  const int b   = blockIdx.x;
  const int tid = threadIdx.x;
  const float* xb = x + (size_t)b * N_ * 3;
  float* db = dist    + (size_t)b * N_;
  float* sb = sampled + (size_t)b * S_ * 3;

  // start index (reference draws it from threefry RNG; fixed to 0 here)
  float cx = xb[0], cy = xb[1], cz = xb[2];
  if (tid == 0) { sb[0] = cx; sb[1] = cy; sb[2] = cz; }

  const int wave = tid >> 5, lane = tid & 31;

  for (int s = 1; s < S_; ++s) {
    float bd = -1.0f; int bi = 0;
    if (s == 1) {
      // first step: dist = min(1e10, d(start)) — initializes the array
      for (int i = tid; i < N_; i += FPS_THREADS) {
        const float* p = xb + 3 * i;
        float dx = p[0] - cx, dy = p[1] - cy, dz = p[2] - cz;
        float dd = fmaf(dx, dx, fmaf(dy, dy, dz * dz));
        dd = fminf(dd, 1e10f);
        db[i] = dd;
        if (dd > bd) { bd = dd; bi = i; }
      }
    } else {
      for (int i = tid; i < N_; i += FPS_THREADS) {
        const float* p = xb + 3 * i;
        float dx = p[0] - cx, dy = p[1] - cy, dz = p[2] - cz;
        float dd = fmaf(dx, dx, fmaf(dy, dy, dz * dz));
        dd = fminf(db[i], dd);
        db[i] = dd;
        if (dd > bd) { bd = dd; bi = i; }
      }
    }
    // wave32 argmax reduction (no barriers)
    #pragma unroll
    for (int off = 16; off > 0; off >>= 1)
      argmax_upd(bd, bi, __shfl_xor(bd, off, 32), __shfl_xor(bi, off, 32));
    if (lane == 0) { sD[wave] = bd; sI[wave] = bi; }
    __syncthreads();
    if (wave == 0) {
      bd = sD[lane]; bi = sI[lane];  // exactly 32 waves -> lane covers all
      #pragma unroll
      for (int off = 16; off > 0; off >>= 1)
        argmax_upd(bd, bi, __shfl_xor(bd, off, 32), __shfl_xor(bi, off, 32));
      if (lane == 0) { sD[0] = bd; sI[0] = bi; }
    }
    __syncthreads();
    const int ci = sI[0];
    __syncthreads();  // protect sD/sI against next iteration's writes
    cx = xb[3 * ci]; cy = xb[3 * ci + 1]; cz = xb[3 * ci + 2];  // broadcast read
    if (tid == 0) { sb[3 * s] = cx; sb[3 * s + 1] = cy; sb[3 * s + 2] = cz; }
  }
}

// ---------------- Layer 1: 3 -> 64, relu ----------------
__global__ __launch_bounds__(256) void mlp1_kernel(
    const float* __restrict__ sampled, const float* __restrict__ W1,
    const float* __restrict__ b1, float* __restrict__ h1) {
  const int g = blockIdx.x * 256 + threadIdx.x;  // B*S*64 threads total
  const int row = g >> 6, c = g & 63;
  const float* p = sampled + 3 * (size_t)row;
  float v = fmaf(p[0], W1[c],
           fmaf(p[1], W1[64 + c],
           fmaf(p[2], W1[128 + c], b1[c])));
  h1[g] = fmaxf(v, 0.0f);
}

// ---- TDM: DMA a full KxNO f32 weight matrix (row-major, contiguous) into LDS ----
#if HAVE_TDM
__device__ __forceinline__ void tdm_load_weights(const float* W, float* sW,
                                                 unsigned int K, unsigned int NO) {
  const unsigned long long ga = (unsigned long long)(uintptr_t)W;
  const unsigned int lds = (unsigned int)(uintptr_t)sW;  // low 32 bits = LDS offset
  v4u g0;
  g0[0] = 1u;                                       // count=1 (valid), user mode
  g0[1] = lds;                                      // lds_addr       bits[63:32]
  g0[2] = (unsigned int)(ga & 0xffffffffu);         // global_addr    bits[95:64]
  g0[3] = (unsigned int)((ga >> 32) & 0x01ffffffu)  // global_addr    bits[120:96]
        | (2u << 30);                               // type=2 (image) bits[127:126]
  const unsigned int dim0 = NO, dim1 = K;           // in data_size (4B) units
  const unsigned long long st0 = NO;                // row stride
  const unsigned long long st1 = (unsigned long long)NO * K;
  v8i g1;
  g1[0] = (int)(2u << 16);                                  // data_size=2 -> 4B
  g1[1] = (int)((dim0 & 0xffffu) << 16);                    // tensor_dim0[15:0]
  g1[2] = (int)(((dim0 >> 16) & 0xffffu) | ((dim1 & 0xffffu) << 16));
  g1[3] = (int)(((dim1 >> 16) & 0xffffu) | ((dim0 & 0xffffu) << 16));  // tile_dim0=dim0
  g1[4] = (int)(dim1 & 0xffffu);                            // tile_dim1=dim1, tile_dim2=0
  g1[5] = (int)(st0 & 0xffffffffull);                       // tensor_dim0_stride[31:0]
  g1[6] = (int)(((st0 >> 32) & 0xffffull) | ((st1 & 0xffffull) << 16));
  g1[7] = (int)((st1 >> 16) & 0xffffffffull);
  v4i g2 = {0, 0, 0, 0};
  v4i g3 = {0, 0, 0, 0};
#if defined(__clang_major__) && __clang_major__ >= 23
  v8i g4 = {0, 0, 0, 0, 0, 0, 0, 0};
  __builtin_amdgcn_tensor_load_to_lds(g0, g1, g2, g3, g4, 0);
#else
  __builtin_amdgcn_tensor_load_to_lds(g0, g1, g2, g3, 0);
#endif
}
#endif

// ---------------- WMMA fp32 GEMM layer: [M x K] @ [K x NO] + bias ----------------
// Per-wave 16-row strip; A frags in VGPRs (lane l: M=l%16, K=4kb+2*(l/16)+{0,1}),
// B frags from LDS (lane l, vgpr j: K=4kb+j+2*(l/16), N=n0+l%16),
// C/D per ISA layout (vgpr j, lane l: M=j+8*(l/16), N=l%16).
template <int K, int NO, bool RELU>
__global__ __launch_bounds__(256) void wmma_layer(
    const float* __restrict__ A, const float* __restrict__ W,
    const float* __restrict__ bias, float* __restrict__ out) {
  extern __shared__ float smem[];
  float* sW = smem;            // K*NO
  float* sB = smem + K * NO;   // NO
  const int tid = threadIdx.x;

#if HAVE_TDM
  if (tid < 32)  // TDM issues per wave (EXEC ignored): launch the DMA once
    tdm_load_weights(W, sW, K, NO);
#else
  for (int i = tid; i < K * NO; i += 256) sW[i] = W[i];
#endif
  for (int i = tid; i < NO; i += 256) sB[i] = bias[i];

  const int wave = tid >> 5, lane = tid & 31;
  const int lr = lane & 15, lh = lane >> 4;
  const int m0 = blockIdx.x * 128 + wave * 16;

  // A fragments can load from global while the TDM DMA is in flight.
  constexpr int KB = K / 4;
  v2f a[KB];
  const float* arow = A + (size_t)(m0 + lr) * K + 2 * lh;
  #pragma unroll
  for (int kb = 0; kb < KB; ++kb) a[kb] = *(const v2f*)(arow + 4 * kb);

#if HAVE_TDM
  __builtin_amdgcn_s_wait_tensorcnt(0);
#endif
  __syncthreads();

  for (int ct = 0; ct < NO / 16; ++ct) {
    const int n0 = ct * 16;
    v8f acc = {};
    #pragma unroll
    for (int kb = 0; kb < KB; ++kb) {
      const float* wp = sW + (size_t)(4 * kb + 2 * lh) * NO + n0 + lr;
      v2f bf;
      bf.x = wp[0];
      bf.y = wp[NO];
      acc = __builtin_amdgcn_wmma_f32_16x16x4_f32(
          /*neg_a=*/false, a[kb], /*neg_b=*/false, bf,
          /*c_mod=*/(short)0, acc, /*reuse_a=*/false, /*reuse_b=*/false);
    }
    const float bv = sB[n0 + lr];
    #pragma unroll
    for (int j = 0; j < 8; ++j) {
      float v = acc[j] + bv;
      if (RELU) v = fmaxf(v, 0.0f);
      out[(size_t)(m0 + j + 8 * lh) * NO + n0 + lr] = v;
    }
  }
}

extern "C" void kernel_launch(void* const* d_in, const int* in_sizes, int n_in,
                              void* d_out, int out_size, void* d_ws, size_t ws_size,
                              hipStream_t stream) {
  (void)in_sizes; (void)n_in; (void)out_size; (void)ws_size;
  const float* x  = (const float*)d_in[0];
  const float* W1 = (const float*)d_in[1];
  const float* b1 = (const float*)d_in[2];
  const float* W2 = (const float*)d_in[3];
  const float* b2 = (const float*)d_in[4];
  const float* W3 = (const float*)d_in[5];
  const float* b3 = (const float*)d_in[6];

  float* out     = (float*)d_out;
  float* sampled = out;                        // B*S*3
  float* feats   = out + (size_t)B_ * S_ * 3;  // B*S*256

  char* ws = (char*)d_ws;
  float* dist = (float*)ws;                                              // B*N f32 (32 MB)
  float* h1   = (float*)(ws + (size_t)B_ * N_ * 4);                      // B*S*64 (16 MB)
  float* h2   = (float*)(ws + (size_t)B_ * N_ * 4 + (size_t)B_ * S_ * 64 * 4);  // B*S*128 (32 MB)

  fps_kernel<<<B_, FPS_THREADS, 0, stream>>>(x, dist, sampled);
  mlp1_kernel<<<(B_ * S_ * 64) / 256, 256, 0, stream>>>(sampled, W1, b1, h1);
  wmma_layer<64, 128, true><<<(B_ * S_) / 128, 256,
      (64 * 128 + 128) * sizeof(float), stream>>>(h1, W2, b2, h2);
  wmma_layer<128, 256, false><<<(B_ * S_) / 128, 256,
      (128 * 256 + 256) * sizeof(float), stream>>>(h2, W3, b3, feats);
}